// SpatialConsistencyLoss_13477607375721
// MI455X (gfx1250) — compile-verified
//
#include <hip/hip_runtime.h>

typedef __attribute__((ext_vector_type(2))) float v2f;
typedef __attribute__((ext_vector_type(8))) float v8f;

#if __has_builtin(__builtin_amdgcn_wmma_f32_16x16x4_f32)
#define USE_WMMA 1
#else
#define USE_WMMA 0
#endif

#define LDS_H 66  // 64 + 2 pad -> lane stride 66 floats => banks (2*j'+h)%64, conflict-free

// ---------------------------------------------------------------------------
// Kernel 1: p[b,i,j] = (1/48) * sum_{c,4x4} (orig - enh)
// Block = 128 threads (4 waves). Each block: 16 pooled rows x 64 pooled cols.
// Grid = 32 batches * 8 rowTiles * 2 colTiles = 512 blocks.
// ---------------------------------------------------------------------------
__global__ __launch_bounds__(128) void scl_pool_diff(
    const float* __restrict__ orig, const float* __restrict__ enh,
    float* __restrict__ p)
{
  __shared__ float y[3][64][LDS_H];  // [channel][local pooled col][input row] W-pooled diff

  const int tid = threadIdx.x;
  const int blk = blockIdx.x;
  const int b       = blk >> 4;
  const int t       = blk & 15;
  const int rowTile = t >> 1;       // 0..7  -> 16 pooled rows
  const int colTile = t & 1;        // 0..1  -> 64 pooled cols
  const int h0 = rowTile * 64;      // input row base
  const int w0 = colTile * 256;     // input col base

  // ---- Phase A: coalesced float4 streaming + W-pooling into LDS ----
  // thread -> fixed local pooled col jp, rows hstart, hstart+2, ...
  const int jp     = tid & 63;
  const int hstart = tid >> 6;      // 0 or 1
  for (int c = 0; c < 3; ++c) {
    const float* ob = orig + (((size_t)b * 3 + c) * 512 + h0) * 512 + w0 + jp * 4;
    const float* eb = enh  + (((size_t)b * 3 + c) * 512 + h0) * 512 + w0 + jp * 4;
    #pragma unroll 4
    for (int h = hstart; h < 64; h += 2) {
      float4 o = *(const float4*)(ob + (size_t)h * 512);
      float4 e = *(const float4*)(eb + (size_t)h * 512);
      y[c][jp][h] = (o.x + o.y + o.z + o.w) - (e.x + e.y + e.z + e.w);
    }
  }
  __syncthreads();

  // ---- Phase B: H + channel pooling as GEMM on the matrix unit ----
  const int lane = tid & 31;
  const int w    = tid >> 5;                 // wave 0..3 -> its own 16x16 tile
  const int nloc = w * 16 + (lane & 15);     // local pooled col for B operand / store
  const int hi   = lane >> 4;                // 0: lanes 0-15, 1: lanes 16-31

#if USE_WMMA
  const int koff = hi << 1;                  // B layout: V0/V1 = K{0,1} or K{2,3}
  v8f acc = {};
  #pragma unroll
  for (int c = 0; c < 3; ++c) {
    #pragma unroll
    for (int kk = 0; kk < 16; ++kk) {
      // A[m,k] = 1/48 iff m == kk (constant pooling matrix slice)
      float a = ((lane & 15) == kk) ? (1.0f / 48.0f) : 0.0f;
      v2f av = {a, a};
      // B[k,n] = y[c][n][4*kk + k]
      const float* src = &y[c][nloc][4 * kk + koff];
      v2f bv = {src[0], src[1]};
      acc = __builtin_amdgcn_wmma_f32_16x16x4_f32(
          false, av, false, bv, (short)0, acc, false, false);
    }
  }
  // D layout: VGPR r -> M = r + 8*hi, N = lane&15
  {
    const int gi = rowTile * 16 + 8 * hi;
    const int gj = colTile * 64 + nloc;
    float* pb = p + ((size_t)b * 128 + gi) * 128 + gj;
    #pragma unroll
    for (int r = 0; r < 8; ++r)
      pb[(size_t)r * 128] = acc[r];
  }
#else
  // VALU fallback: same output mapping
  {
    const int gj = colTile * 64 + nloc;
    #pragma unroll
    for (int r = 0; r < 8; ++r) {
      int m = 8 * hi + r;
      float s = 0.0f;
      #pragma unroll
      for (int c = 0; c < 3; ++c)
        #pragma unroll
        for (int k = 0; k < 4; ++k)
          s += y[c][nloc][4 * m + k];
      p[((size_t)b * 128 + rowTile * 16 + m) * 128 + gj] = s * (1.0f / 48.0f);
    }
  }
#endif
}

// ---------------------------------------------------------------------------
// Kernel 2: loss = sum over 4 directions of (p - neighbor)^2, zero padding.
// p is 2 MB -> L2 resident; trivially cheap.
// ---------------------------------------------------------------------------
__global__ __launch_bounds__(256) void scl_loss(
    const float* __restrict__ p, float* __restrict__ out)
{
  const int idx = blockIdx.x * 256 + threadIdx.x;   // 32*128*128 threads
  const int j = idx & 127;
  const int i = (idx >> 7) & 127;
  const float c = p[idx];
  const float l = (j > 0)   ? p[idx - 1]   : 0.0f;
  const float r = (j < 127) ? p[idx + 1]   : 0.0f;
  const float u = (i > 0)   ? p[idx - 128] : 0.0f;
  const float d = (i < 127) ? p[idx + 128] : 0.0f;
  const float dl = c - l, dr = c - r, du = c - u, dd = c - d;
  out[idx] = dl * dl + dr * dr + du * du + dd * dd;
}

extern "C" void kernel_launch(void* const* d_in, const int* in_sizes, int n_in,
                              void* d_out, int out_size, void* d_ws, size_t ws_size,
                              hipStream_t stream) {
  const float* orig = (const float*)d_in[0];   // [32,3,512,512] f32
  const float* enh  = (const float*)d_in[1];   // [32,3,512,512] f32
  float* out = (float*)d_out;                  // [32,1,128,128] f32
  float* p   = (float*)d_ws;                   // 32*128*128 floats = 2 MB scratch

  scl_pool_diff<<<dim3(512), dim3(128), 0, stream>>>(orig, enh, p);
  scl_loss<<<dim3(32 * 128 * 128 / 256), dim3(256), 0, stream>>>(p, out);
}